// DualChannelRelationWeights_46883863003178
// MI455X (gfx1250) — compile-verified
//
#include <hip/hip_runtime.h>
#include <hip/hip_bf16.h>
#include <math.h>

typedef float v2f __attribute__((ext_vector_type(2)));
typedef float v8f __attribute__((ext_vector_type(8)));

#define N_NODES   100000
#define PROJ_DIM  128
#define GAT_DIM   64
#define NEG_SLOPE 0.2f

// Workspace layout (in floats):
//   [0   .. 511]  vtab[4][128] : v1b, v2b, v1p, v2p   (v = W^T * a_half)
//   [512 .. 513]  softmax weights w0, w1
//   [640 .. 640+4N) per-node table [N][4] : si_b, sj_b, si_p, sj_p (16B aligned)
#define WS_VTAB 0
#define WS_W    512
#define WS_TAB  640

// ---------------------------------------------------------------------------
// Kernel 1: fold W and a into 4 vectors of length 128; compute softmax(gamma).
// One block, 128 threads (one per k). Trivial cost.
// ---------------------------------------------------------------------------
__global__ void prep_kernel(const float* __restrict__ Wb, const float* __restrict__ Wp,
                            const float* __restrict__ ab, const float* __restrict__ ap,
                            const float* __restrict__ gb, const float* __restrict__ gp,
                            float* __restrict__ ws) {
    int k = threadIdx.x;  // 0..127
    float v1b = 0.f, v2b = 0.f, v1p = 0.f, v2p = 0.f;
    #pragma unroll 4
    for (int d = 0; d < GAT_DIM; ++d) {
        float wbv = Wb[d * PROJ_DIM + k];
        float wpv = Wp[d * PROJ_DIM + k];
        v1b = fmaf(wbv, ab[d], v1b);
        v2b = fmaf(wbv, ab[GAT_DIM + d], v2b);
        v1p = fmaf(wpv, ap[d], v1p);
        v2p = fmaf(wpv, ap[GAT_DIM + d], v2p);
    }
    ws[WS_VTAB + 0 * PROJ_DIM + k] = v1b;
    ws[WS_VTAB + 1 * PROJ_DIM + k] = v2b;
    ws[WS_VTAB + 2 * PROJ_DIM + k] = v1p;
    ws[WS_VTAB + 3 * PROJ_DIM + k] = v2p;
    if (k == 0) {
        // softmax over [gamma_b, gamma_p]
        float w0 = 1.0f / (1.0f + expf(gp[0] - gb[0]));
        ws[WS_W + 0] = w0;
        ws[WS_W + 1] = 1.0f - w0;
    }
}

// ---------------------------------------------------------------------------
// Kernel 2: node phase. [N,128] @ [128,2] per channel via V_WMMA_F32_16X16X4_F32.
// One wave32 per 16-node tile; 32 K-steps of 2 WMMAs each (channel b, channel p).
// A: lane L holds z[row=L&15][k0+koff .. +1], koff = (L>=16)?2:0 (ISA 16x4 f32 layout)
// B: lane L holds column n=L&15; columns >=2 are routed at a zero-padded LDS row
//    so every lane issues an unconditional ds load (no EXEC churn in the loop).
// D: column n lives in lane n (M=0..7 in VGPR 0..7) and lane n+16 (M=8..15).
// ---------------------------------------------------------------------------
__global__ void node_kernel(const float* __restrict__ zb, const float* __restrict__ zp,
                            float* __restrict__ ws) {
    // Rows 0..3: v1b, v2b, v1p, v2p.  Rows 4..7: zeros (dead-column target).
    __shared__ float vt[8 * PROJ_DIM];
    int t = threadIdx.x;                     // 0..255 (8 waves)
    vt[t]        = ws[WS_VTAB + t];
    vt[t + 256]  = ws[WS_VTAB + t + 256];
    vt[t + 512]  = 0.f;
    vt[t + 768]  = 0.f;
    __syncthreads();

    int wave = t >> 5;
    int lane = t & 31;
    int tile = blockIdx.x * 8 + wave;        // wave-uniform
    int row0 = tile * 16;
    if (row0 >= N_NODES) return;             // uniform across the wave

    int m    = lane & 15;                    // A-row / B-column index
    int koff = (lane >> 4) << 1;             // 0 or 2
    int row  = row0 + m;
    if (row >= N_NODES) row = N_NODES - 1;   // safety clamp (N%16==0 here)

    const float* arb = zb + (size_t)row * PROJ_DIM + koff;
    const float* arp = zp + (size_t)row * PROJ_DIM + koff;

    // B base: live columns (m<2) read rows m (b) / m+2 (p); dead columns read
    // zero rows 4 (b) / 6 (p). The b->p delta is a constant 1024 bytes for all
    // lanes, so both loads can fuse into one ds_load_2addr.
    int selb = (m < 2) ? m : 4;
    const float* vbase = &vt[selb * PROJ_DIM + koff];

    v8f cb = {}; v8f cp = {};
    #pragma unroll
    for (int k0 = 0; k0 < PROJ_DIM; k0 += 4) {
        v2f aB, aP, bB, bP;
        aB.x = arb[k0];       aB.y = arb[k0 + 1];
        aP.x = arp[k0];       aP.y = arp[k0 + 1];
        bB.x = vbase[k0];     bB.y = vbase[k0 + 1];                 // row selb
        bP.x = vbase[k0 + 2 * PROJ_DIM];                            // row selb+2
        bP.y = vbase[k0 + 2 * PROJ_DIM + 1];
        cb = __builtin_amdgcn_wmma_f32_16x16x4_f32(false, aB, false, bB,
                                                   (short)0, cb, false, false);
        cp = __builtin_amdgcn_wmma_f32_16x16x4_f32(false, aP, false, bP,
                                                   (short)0, cp, false, false);
    }

    // Scatter the two live columns (n=0: si, n=1: sj) of each accumulator.
    if (m < 2) {
        int mbase = row0 + (lane >> 4) * 8;  // VGPR r holds M = (lane>=16?8:0)+r
        float* tab = ws + WS_TAB;
        #pragma unroll
        for (int r = 0; r < 8; ++r) {
            int node = mbase + r;
            if (node < N_NODES) {
                tab[(size_t)node * 4 + m]     = cb[r];   // channel b
                tab[(size_t)node * 4 + 2 + m] = cp[r];   // channel p
            }
        }
    }
}

// ---------------------------------------------------------------------------
// Kernel 3: edge phase. Gather 16B/node, symmetric leaky-relu score, sigmoid.
// ---------------------------------------------------------------------------
__device__ __forceinline__ float lrelu(float x) {
    return x > 0.f ? x : NEG_SLOPE * x;
}

__global__ void edge_kernel(const int* __restrict__ ei, const float* __restrict__ ws,
                            float* __restrict__ out, int E) {
    int e = blockIdx.x * blockDim.x + threadIdx.x;
    if (e >= E) return;
    int src = ei[e];
    int dst = ei[E + e];
    const float4* tab = (const float4*)(ws + WS_TAB);
    float4 ts = tab[src];   // {si_b, sj_b, si_p, sj_p}
    float4 td = tab[dst];
    float w0 = ws[WS_W], w1 = ws[WS_W + 1];
    float sb = 0.5f * (lrelu(ts.x + td.y) + lrelu(td.x + ts.y));
    float sp = 0.5f * (lrelu(ts.z + td.w) + lrelu(td.z + ts.w));
    float s  = w0 * sb + w1 * sp;           // TAU_MRF == 1
    out[e] = 1.0f / (1.0f + expf(-s));
}

extern "C" void kernel_launch(void* const* d_in, const int* in_sizes, int n_in,
                              void* d_out, int out_size, void* d_ws, size_t ws_size,
                              hipStream_t stream) {
    const float* zb = (const float*)d_in[0];
    const float* zp = (const float*)d_in[1];
    const float* Wb = (const float*)d_in[2];
    const float* Wp = (const float*)d_in[3];
    const float* ab = (const float*)d_in[4];
    const float* ap = (const float*)d_in[5];
    const float* gb = (const float*)d_in[6];
    const float* gp = (const float*)d_in[7];
    const int*   ei = (const int*)d_in[8];
    float* ws  = (float*)d_ws;
    float* out = (float*)d_out;
    int E = in_sizes[8] / 2;

    prep_kernel<<<1, 128, 0, stream>>>(Wb, Wp, ab, ap, gb, gp, ws);

    int tiles  = (N_NODES + 15) / 16;        // 6250 wave-tiles
    int blocks = (tiles + 7) / 8;            // 8 waves (256 thr) per block
    node_kernel<<<blocks, 256, 0, stream>>>(zb, zp, ws);

    edge_kernel<<<(E + 255) / 256, 256, 0, stream>>>(ei, ws, out, E);
}